// GraphGAE_6983616823296
// MI455X (gfx1250) — compile-verified
//
#include <hip/hip_runtime.h>

#define T_STEPS 512
#define NNODES  512
#define NEDGES  16384
#define DIM     256
#define HID     256
#define HC      32            // GEMM1 column chunk
#define NCHUNK  (HID / HC)    // 8
#define GAMMA_F 0.99f
#define LMBDA_F 0.95f
#define HPAD    33            // padded row stride for h chunk in LDS
#define WS1     40            // GEMM1 W tile row stride in dwords (32 + 8 pad)
#define WS2     72            // GEMM2 W tile row stride in dwords (64 + 8 pad)
#define NTHREADS 512

typedef float v2f __attribute__((ext_vector_type(2)));
typedef float v8f __attribute__((ext_vector_type(8)));
typedef unsigned v4u __attribute__((ext_vector_type(4)));
typedef int v8i __attribute__((ext_vector_type(8)));
typedef int v4i __attribute__((ext_vector_type(4)));

// LDS layout (floats), dynamic shared starts at LDS offset 0 (no static LDS):
//   hbuf : NNODES*HPAD  (16896)   h chunk, padded rows
//   aggl : NNODES*HC    (16384)   aggregation chunk
//   wbuf : DIM*WS2      (18432)   k-major weight tile, TDM-padded rows
//   csum : HID          (  256)   column sums of relu(h2)
#define HBUF_F 0
#define AGGL_F (NNODES * HPAD)
#define WBUF_F (AGGL_F + NNODES * HC)
#define CSUM_F (WBUF_F + DIM * WS2)
#define LDS_FLOATS (CSUM_F + HID)
#define WBUF_BYTE_OFF (WBUF_F * 4)

// D = A(16x4 f32) * B(4x16 f32) + C(16x16 f32), wave32 striped
__device__ __forceinline__ v8f wmma_f32(v2f a, v2f b, v8f c) {
  return __builtin_amdgcn_wmma_f32_16x16x4_f32(false, a, false, b, (short)0, c,
                                               false, false);
}

// Tensor Data Mover: DMA a 2-D tile (tileW x tileH elems, f32) from global
// (row stride strideElems) into LDS at ldsByteOff, inserting coded pad
// dwords of LDS padding every coded interval.  D# per ISA ch.8.
__device__ __forceinline__ void tdm_load_2d(unsigned ldsByteOff,
                                            const void* gptr,
                                            unsigned tensorW, unsigned tensorH,
                                            unsigned tileW, unsigned tileH,
                                            unsigned strideElems,
                                            unsigned padIntCode,
                                            unsigned padAmtCode) {
  unsigned long long ga = (unsigned long long)(uintptr_t)gptr;
  v4u g0;
  g0[0] = 1u;                                    // count=1, user descriptor
  g0[1] = ldsByteOff;                            // lds_addr
  g0[2] = (unsigned)(ga & 0xffffffffu);          // global_addr[31:0]
  g0[3] = (unsigned)((ga >> 32) & 0x01ffffffu)   // global_addr[56:32]
        | (2u << 30);                            // type = 2 ("image")
  v8i g1;
  g1[0] = (int)((2u << 16)                       // data_size: 4 bytes
              | (1u << 20)                       // pad_enable
              | (padIntCode << 22)               // pad_interval
              | (padAmtCode << 25));             // pad_amount
  g1[1] = (int)((tensorW & 0xffffu) << 16);      // tensor_dim0[15:0]
  g1[2] = (int)(((tensorW >> 16) & 0xffffu) | ((tensorH & 0xffffu) << 16));
  g1[3] = (int)(((tensorH >> 16) & 0xffffu) | ((tileW & 0xffffu) << 16));
  g1[4] = (int)(tileH & 0xffffu);                // tile_dim1 (tile_dim2 = 0)
  g1[5] = (int)strideElems;                      // tensor_dim0_stride[31:0]
  g1[6] = 0;
  g1[7] = 0;
  v4i z4 = {0, 0, 0, 0};
  v8i z8 = {0, 0, 0, 0, 0, 0, 0, 0};
  __builtin_amdgcn_tensor_load_to_lds(g0, g1, z4, z4, z8, 0);
}

__global__ void __launch_bounds__(NTHREADS)
gnn_value_kernel(const float* __restrict__ nodeF,
                 const float* __restrict__ nextF,
                 const int* __restrict__ eiCur,
                 const int* __restrict__ eiNext,
                 const float* __restrict__ pW1,
                 const float* __restrict__ pb1,
                 const float* __restrict__ pW2,
                 const float* __restrict__ pb2,
                 const float* __restrict__ pwout,
                 const float* __restrict__ pbout,
                 float* __restrict__ aggBase,
                 float* __restrict__ values) {
  extern __shared__ float smem[];
  float* hbuf = smem + HBUF_F;   // [NNODES][HPAD]
  float* aggl = smem + AGGL_F;   // [NNODES][HC]
  float* wbuf = smem + WBUF_F;   // k-major weight tile, row stride WS1/WS2
  float* csum = smem + CSUM_F;   // [HID]

  const int tid  = threadIdx.x;
  const int lane = tid & 31;
  const int wv   = tid >> 5;      // 0..15
  const int m    = lane & 15;     // M (or N) index within a 16-wide tile
  const int kh   = lane >> 4;     // which K-pair half this lane feeds

  float* aggSlab = aggBase + (size_t)blockIdx.x * NNODES * HID;

  for (int work = blockIdx.x; work < 2 * T_STEPS; work += gridDim.x) {
    const int t = work >> 1;
    const int side = work & 1;
    const float* x  = (side ? nextF : nodeF) + (size_t)t * NNODES * DIM;
    const int*   e0 = (side ? eiNext : eiCur) + (size_t)t * 2 * NEDGES;
    const int*   e1 = e0 + NEDGES;

    // Warm L2 for the next work item's feature tile while we compute.
    {
      int nwork = work + gridDim.x;
      if (nwork < 2 * T_STEPS) {
        const float* xn = ((nwork & 1) ? nextF : nodeF) +
                          (size_t)(nwork >> 1) * NNODES * DIM;
        __builtin_prefetch(xn + (size_t)tid * 64, 0, 0);
      }
    }

    // ================= Stage A: h = relu(x@W1+b1), segment_sum -> agg ======
    for (int c = 0; c < NCHUNK; ++c) {
      const int c0 = c * HC;
      // TDM: stage W1[:, c0:c0+HC] k-major into wbuf with 8-dword row pad.
      if (wv == 0)
        tdm_load_2d(WBUF_BYTE_OFF, pW1 + c0, HID, DIM, HC, DIM, HID,
                    /*pad every 32 dw*/ 4u, /*pad 8 dw*/ 7u);
      // Everyone zeros the aggregation chunk while the DMA runs.
      for (int idx = tid; idx < NNODES * HC; idx += NTHREADS) aggl[idx] = 0.0f;
      if (wv == 0) __builtin_amdgcn_s_wait_tensorcnt(0);
      __syncthreads();

      // GEMM1: each wave does 2 row tiles x 2 col tiles of this chunk
      for (int rt = 0; rt < 2; ++rt) {
        const int m0 = (wv * 2 + rt) * 16;
        v8f acc0 = {};
        v8f acc1 = {};
        const float* arow = x + (size_t)(m0 + m) * DIM + 2 * kh;
        const float* bp0  = wbuf + 2 * kh * WS1 + 0 * 16 + m;  // W[k][n]
        const float* bp1  = wbuf + 2 * kh * WS1 + 1 * 16 + m;
#pragma unroll 4
        for (int k0 = 0; k0 < DIM; k0 += 4) {
          v2f a = *(const v2f*)(arow + k0);
          v2f b0, b1v;
          b0.x  = bp0[k0 * WS1];
          b0.y  = bp0[k0 * WS1 + WS1];
          b1v.x = bp1[k0 * WS1];
          b1v.y = bp1[k0 * WS1 + WS1];
          acc0 = wmma_f32(a, b0, acc0);
          acc1 = wmma_f32(a, b1v, acc1);
        }
        const float bias0 = pb1[c0 + m];
        const float bias1 = pb1[c0 + 16 + m];
#pragma unroll
        for (int r = 0; r < 8; ++r) {
          int row = m0 + r + 8 * kh;            // C layout: M = r + 8*(lane>=16)
          float h0 = acc0[r] + bias0;
          float h1 = acc1[r] + bias1;
          hbuf[row * HPAD + m]      = h0 > 0.0f ? h0 : 0.0f;
          hbuf[row * HPAD + 16 + m] = h1 > 0.0f ? h1 : 0.0f;
        }
      }
      __syncthreads();

      // Edge aggregation: agg[dst,:] += h[src,:] via LDS float atomics
      for (int e = wv; e < NEDGES; e += 16) {
        int s = e0[e];
        int d = e1[e];
        float v = hbuf[s * HPAD + lane];
        atomicAdd(&aggl[d * HC + lane], v);
      }
      __syncthreads();

      // Coalesced writeback of this agg chunk to the block's scratch slab
      for (int idx = tid; idx < NNODES * HC; idx += NTHREADS) {
        int row = idx >> 5;
        int col = idx & (HC - 1);
        aggSlab[(size_t)row * HID + c0 + col] = aggl[idx];
      }
      __syncthreads();
    }
    __threadfence();   // make agg slab stores visible to all waves in block
    __syncthreads();

    // ========== Stage B: colsum = sum_n relu(agg@W2+b2), value out =========
    for (int idx = tid; idx < HID; idx += NTHREADS) csum[idx] = 0.0f;
    __syncthreads();

    for (int g = 0; g < 4; ++g) {
      const int n0 = g * 64;
      // TDM: stage W2[:, n0:n0+64] k-major into wbuf with 8-dword row pad.
      if (wv == 0) {
        tdm_load_2d(WBUF_BYTE_OFF, pW2 + n0, HID, HID, 64, HID, HID,
                    /*pad every 64 dw*/ 5u, /*pad 8 dw*/ 7u);
        __builtin_amdgcn_s_wait_tensorcnt(0);
      }
      __syncthreads();

      for (int rt = 0; rt < 2; ++rt) {
        const int m0 = (wv * 2 + rt) * 16;
        v8f accs[4] = {{}, {}, {}, {}};
        const float* arow = aggSlab + (size_t)(m0 + m) * HID + 2 * kh;
        const float* bp   = wbuf + 2 * kh * WS2 + m;
#pragma unroll 2
        for (int k0 = 0; k0 < HID; k0 += 4) {
          v2f a = *(const v2f*)(arow + k0);
#pragma unroll
          for (int nt = 0; nt < 4; ++nt) {
            v2f b;
            b.x = bp[k0 * WS2 + nt * 16];
            b.y = bp[k0 * WS2 + WS2 + nt * 16];
            accs[nt] = wmma_f32(a, b, accs[nt]);
          }
        }
#pragma unroll
        for (int nt = 0; nt < 4; ++nt) {
          float bias = pb2[n0 + nt * 16 + m];
          float s = 0.0f;
#pragma unroll
          for (int r = 0; r < 8; ++r) {
            float hv = accs[nt][r] + bias;
            s += hv > 0.0f ? hv : 0.0f;
          }
          atomicAdd(&csum[n0 + nt * 16 + m], s);
        }
      }
      __syncthreads();
    }

    if (tid < 32) {
      float p = 0.0f;
      for (int n = lane; n < HID; n += 32) p += csum[n] * pwout[n];
#pragma unroll
      for (int off = 16; off; off >>= 1) p += __shfl_down(p, off);
      if (lane == 0)
        values[side * T_STEPS + t] = p * (1.0f / NNODES) + pbout[0];
    }
    __syncthreads();
  }
}

// Sequential reverse GAE scan (T=512, trivial cost)
__global__ void gae_scan_kernel(const float* __restrict__ values,
                                const float* __restrict__ rewards,
                                const unsigned char* __restrict__ dones,
                                float* __restrict__ out) {
  if (threadIdx.x != 0 || blockIdx.x != 0) return;
  float gae = 0.0f;
  for (int t = T_STEPS - 1; t >= 0; --t) {
    float nd = dones[t] ? 0.0f : 1.0f;
    float v  = values[t];
    float nv = values[T_STEPS + t] * nd;
    float td = rewards[t] + GAMMA_F * nv - v;
    gae = td + GAMMA_F * LMBDA_F * gae * nd;
    out[2 * t]     = gae;       // advantage
    out[2 * t + 1] = gae + v;   // value target
  }
}

extern "C" void kernel_launch(void* const* d_in, const int* in_sizes, int n_in,
                              void* d_out, int out_size, void* d_ws,
                              size_t ws_size, hipStream_t stream) {
  (void)in_sizes; (void)n_in; (void)out_size;
  const float* nodeF = (const float*)d_in[0];
  const float* nextF = (const float*)d_in[1];
  const int*   ei    = (const int*)d_in[2];
  const int*   nei   = (const int*)d_in[3];
  const float* rew   = (const float*)d_in[4];
  const unsigned char* dones = (const unsigned char*)d_in[5];
  const float* W1    = (const float*)d_in[6];
  const float* b1    = (const float*)d_in[7];
  const float* W2    = (const float*)d_in[8];
  const float* b2    = (const float*)d_in[9];
  const float* wout  = (const float*)d_in[10];
  const float* bout  = (const float*)d_in[11];

  float* values  = (float*)d_ws;                       // 2*T floats
  float* aggBase = (float*)((char*)d_ws + 4096);
  const size_t slabBytes = (size_t)NNODES * HID * sizeof(float);
  int slabs = 1;
  if (ws_size > 4096 + slabBytes)
    slabs = (int)((ws_size - 4096) / slabBytes);
  if (slabs < 1) slabs = 1;
  if (slabs > 256) slabs = 256;          // <=128MB scratch, L2-resident

  const size_t ldsBytes = (size_t)LDS_FLOATS * sizeof(float);  // ~203 KB
  (void)hipFuncSetAttribute((const void*)gnn_value_kernel,
                            hipFuncAttributeMaxDynamicSharedMemorySize,
                            (int)ldsBytes);

  gnn_value_kernel<<<slabs, NTHREADS, ldsBytes, stream>>>(
      nodeF, nextF, ei, nei, W1, b1, W2, b2, wout, bout, aggBase, values);
  gae_scan_kernel<<<1, 32, 0, stream>>>(values, rew, dones, (float*)d_out);
}